// EddyFormer_86973087744455
// MI455X (gfx1250) — compile-verified
//
#include <hip/hip_runtime.h>
#include <cmath>

// ---------------------------------------------------------------------------
// EddyFormer forward for MI455X (gfx1250, wave32, WMMA).
// All dense GEMMs use V_WMMA_F32_16X16X4_F32 (native fp32 matrix pipe).
// Round 3: fragment-packed weights (1x b64 per B-frag), 2 M-tiles per wave,
// depth-1 software pipelining so global-load latency overlaps the WMMA pipe.
// ---------------------------------------------------------------------------

typedef float v2f __attribute__((ext_vector_type(2)));
typedef float v8f __attribute__((ext_vector_type(8)));

#define NE    512          // MESH^3 elements
#define P8_3  512          // 8^3 points (sgs)
#define P4_3  64           // 4^3 points (les)
#define CCH   64
#define FFH   256
#define NHEAD 4
#define DHEAD 32
#define NLAY  2
#define TOK_S (NE * P8_3)  // 262144 sgs tokens
#define TOK_L (NE * P4_3)  // 32768  les tokens

// ---- workspace float offsets --------------------------------------------
#define MOFF_C8    0
#define MOFF_C4    8
#define MOFF_FG    16     // 8x8
#define MOFF_EV    80     // 8x8
#define MOFF_DN    144    // 4x8
#define MOFF_UP    176    // 8x4
#define MOFF_C2M8  208    // 8x8
#define MOFF_M2N8  272    // 8x8
#define MOFF_C2M4  336    // 4x4
#define MOFF_M2N4  352    // 4x4
#define MATS_F     1024

static __device__ __forceinline__ v8f wmma4(v2f a, v2f b, v8f c) {
  // D = A(16x4,f32) * B(4x16,f32) + C(16x16,f32)
  return __builtin_amdgcn_wmma_f32_16x16x4_f32(false, a, false, b, (short)0, c,
                                               false, false);
}

static __device__ __forceinline__ float gelu_tanh(float v) {
  float u = 0.7978845608028654f * (v + 0.044715f * v * v * v);
  return 0.5f * v * (1.0f + tanhf(u));
}

// ===========================================================================
// 0. Build Chebyshev operator matrices on device (1 thread, doubles).
// ===========================================================================
__device__ void d_chebvander(const double* x, int npts, int cols, double* V) {
  for (int i = 0; i < npts; ++i) {
    double tm1 = 1.0, tc = x[i];
    for (int j = 0; j < cols; ++j) {
      double t;
      if (j == 0)      t = 1.0;
      else if (j == 1) t = x[i];
      else { t = 2.0 * x[i] * tc - tm1; tm1 = tc; tc = t; }
      V[i * cols + j] = t;
    }
  }
}

__device__ void d_minv(const double* A, double* Inv, int n) {
  double M[8][16];
  for (int i = 0; i < n; ++i)
    for (int j = 0; j < n; ++j) { M[i][j] = A[i*n+j]; M[i][n+j] = (i == j) ? 1.0 : 0.0; }
  for (int c = 0; c < n; ++c) {
    int piv = c;
    for (int r = c + 1; r < n; ++r) if (fabs(M[r][c]) > fabs(M[piv][c])) piv = r;
    if (piv != c) for (int j = 0; j < 2*n; ++j) { double t = M[c][j]; M[c][j] = M[piv][j]; M[piv][j] = t; }
    double inv = 1.0 / M[c][c];
    for (int j = 0; j < 2*n; ++j) M[c][j] *= inv;
    for (int r = 0; r < n; ++r) if (r != c) {
      double f = M[r][c];
      for (int j = 0; j < 2*n; ++j) M[r][j] -= f * M[c][j];
    }
  }
  for (int i = 0; i < n; ++i)
    for (int j = 0; j < n; ++j) Inv[i*n+j] = M[i][n+j];
}

__device__ void d_mmulf(const double* A, const double* B, float* C, int m, int k, int n) {
  for (int i = 0; i < m; ++i)
    for (int j = 0; j < n; ++j) {
      double s = 0.0;
      for (int q = 0; q < k; ++q) s += A[i*k+q] * B[q*n+j];
      C[i*n+j] = (float)s;
    }
}

__global__ void k_init_mats(float* mats) {
  if (threadIdx.x != 0 || blockIdx.x != 0) return;
  const double PI = 3.14159265358979323846;
  double c8[8], c4[4], u8[8];
  for (int i = 0; i < 8; ++i) c8[i] = -cos(PI * i / 7.0);
  for (int i = 0; i < 4; ++i) c4[i] = -cos(PI * i / 3.0);
  for (int i = 0; i < 8; ++i) u8[i] = ((i + 0.5) / 8.0) * 2.0 - 1.0;
  double V8[64], Vu8[64], V4[16], W48[32], W84[32];
  d_chebvander(c8, 8, 8, V8);
  d_chebvander(u8, 8, 8, Vu8);
  d_chebvander(c4, 4, 4, V4);
  d_chebvander(c4, 4, 8, W48);
  d_chebvander(c8, 8, 4, W84);
  double iV8[64], iVu8[64], iV4[16];
  d_minv(V8, iV8, 8);
  d_minv(Vu8, iVu8, 8);
  d_minv(V4, iV4, 4);
  d_mmulf(V8,  iVu8, mats + MOFF_FG, 8, 8, 8);   // FG: u8 nodal -> c8 nodal
  d_mmulf(Vu8, iV8,  mats + MOFF_EV, 8, 8, 8);   // EV: c8 nodal -> u8 nodal
  d_mmulf(W48, iV8,  mats + MOFF_DN, 4, 8, 8);   // DN: c8 -> c4   (4x8)
  d_mmulf(W84, iV4,  mats + MOFF_UP, 8, 4, 4);   // UP: c4 -> c8   (8x4)
  for (int i = 0; i < 64; ++i) { mats[MOFF_C2M8+i] = (float)iV8[i]; mats[MOFF_M2N8+i] = (float)V8[i]; }
  for (int i = 0; i < 16; ++i) { mats[MOFF_C2M4+i] = (float)iV4[i]; mats[MOFF_M2N4+i] = (float)V4[i]; }
  for (int i = 0; i < 8; ++i) mats[MOFF_C8+i] = (float)c8[i];
  for (int i = 0; i < 4; ++i) mats[MOFF_C4+i] = (float)c4[i];
}

// ===========================================================================
// 0b. Pack a KxN weight into fragment order: Bp[p][n] = {B[2p][n], B[2p+1][n]}
// ===========================================================================
__global__ void k_packB(const float* __restrict__ B, float* __restrict__ Bp,
                        int K, int N) {
  int gid = blockIdx.x * 256 + threadIdx.x;
  int total = (K / 2) * N;
  if (gid >= total) return;
  int p = gid / N, n = gid % N;
  Bp[(size_t)gid * 2 + 0] = B[(size_t)(2 * p)     * N + n];
  Bp[(size_t)gid * 2 + 1] = B[(size_t)(2 * p + 1) * N + n];
}

// ===========================================================================
// 1. Embed: gather element tile, apply3(FG), append coords -> xg7 (NE,512,7)
// ===========================================================================
__global__ void k_embed(const float* __restrict__ in, float* __restrict__ xg7,
                        const float* __restrict__ mats) {
  __shared__ float bufA[512 * 4];
  __shared__ float bufB[512 * 4];
  const float* FG = mats + MOFF_FG;
  const float* c8 = mats + MOFF_C8;
  const int tid = threadIdx.x;
  const int e  = blockIdx.x;
  const int mi = e >> 6, mj = (e >> 3) & 7, mk = e & 7;
  for (int t = tid; t < 512 * 4; t += 256) {
    int pt = t >> 2, c = t & 3;
    int i = pt >> 6, j = (pt >> 3) & 7, k = pt & 7;
    bufA[t] = in[(((size_t)(mi*8+i) * 64 + (mj*8+j)) * 64 + (mk*8+k)) * 4 + c];
  }
  __syncthreads();
  for (int t = tid; t < 512 * 4; t += 256) {   // contract i
    int pt = t >> 2, c = t & 3;
    int a = pt >> 6, rem = pt & 63;
    float s = 0.f;
    for (int q = 0; q < 8; ++q) s += FG[a*8+q] * bufA[((q << 6) | rem) * 4 + c];
    bufB[t] = s;
  }
  __syncthreads();
  for (int t = tid; t < 512 * 4; t += 256) {   // contract j
    int pt = t >> 2, c = t & 3;
    int k = pt & 7, ab = pt >> 3;
    int b = ab & 7, a = ab >> 3;
    float s = 0.f;
    for (int q = 0; q < 8; ++q) s += FG[b*8+q] * bufB[((a << 6) | (q << 3) | k) * 4 + c];
    bufA[t] = s;
  }
  __syncthreads();
  for (int t = tid; t < 512 * 4; t += 256) {   // contract k
    int pt = t >> 2, c = t & 3;
    int cp = pt & 7, ab = pt >> 3;
    float s = 0.f;
    for (int q = 0; q < 8; ++q) s += FG[cp*8+q] * bufA[((ab << 3) | q) * 4 + c];
    xg7[((size_t)e * 512 + pt) * 7 + c] = s;
  }
  const float hh = 0.78539816339744830962f;  // 2*pi/8
  for (int pt = tid; pt < 512; pt += 256) {
    int i = pt >> 6, j = (pt >> 3) & 7, k = pt & 7;
    size_t base = ((size_t)e * 512 + pt) * 7;
    xg7[base + 4] = hh * ((float)mi + 0.5f * (c8[i] + 1.0f));
    xg7[base + 5] = hh * ((float)mj + 0.5f * (c8[j] + 1.0f));
    xg7[base + 6] = hh * ((float)mk + 0.5f * (c8[k] + 1.0f));
  }
}

// ===========================================================================
// 2. Generic per-element separable transform: y = apply3(x, mat,mat,mat)
// ===========================================================================
__global__ void k_apply3(const float* __restrict__ x, float* __restrict__ y,
                         const float* __restrict__ mat, int Pin, int Pout, int C,
                         const float* __restrict__ scale, int accumulate) {
  __shared__ float bufA[512 * 8];
  __shared__ float bufB[512 * 8];
  const int tid = threadIdx.x;
  const int e = blockIdx.x;
  const int Pin2 = Pin * Pin, Pin3 = Pin2 * Pin;
  const int Pout3 = Pout * Pout * Pout;
  for (int c0 = 0; c0 < C; c0 += 8) {
    const int cc = (C - c0 < 8) ? (C - c0) : 8;
    for (int t = tid; t < Pin3 * cc; t += 256) {
      int pt = t / cc, c = t % cc;
      bufA[pt * 8 + c] = x[((size_t)e * Pin3 + pt) * C + c0 + c];
    }
    __syncthreads();
    const int n1 = Pout * Pin2;
    for (int t = tid; t < n1 * cc; t += 256) {
      int pt = t / cc, c = t % cc;
      int a = pt / Pin2, rem = pt % Pin2;
      float s = 0.f;
      for (int q = 0; q < Pin; ++q) s += mat[a*Pin+q] * bufA[(q*Pin2 + rem) * 8 + c];
      bufB[pt * 8 + c] = s;          // (a,j,k)
    }
    __syncthreads();
    const int n2 = Pout * Pout * Pin;
    for (int t = tid; t < n2 * cc; t += 256) {
      int pt = t / cc, c = t % cc;
      int k = pt % Pin, ab = pt / Pin;
      int b = ab % Pout, a = ab / Pout;
      float s = 0.f;
      for (int q = 0; q < Pin; ++q) s += mat[b*Pin+q] * bufB[(a*Pin2 + q*Pin + k) * 8 + c];
      bufA[pt * 8 + c] = s;          // (a,b,k)
    }
    __syncthreads();
    for (int t = tid; t < Pout3 * cc; t += 256) {
      int pt = t / cc, c = t % cc;
      int cp = pt % Pout, ab = pt / Pout;
      float s = 0.f;
      for (int q = 0; q < Pin; ++q) s += mat[cp*Pin+q] * bufA[(ab*Pin + q) * 8 + c];
      if (scale) s *= scale[c0 + c];
      size_t oi = ((size_t)e * Pout3 + pt) * C + c0 + c;
      if (accumulate) y[oi] += s; else y[oi] = s;
    }
    __syncthreads();
  }
}

// ===========================================================================
// 3. Lift: (tokens x 7) @ (7 x 64) + b   (tiny K -> VALU)
// ===========================================================================
__global__ void k_lift(const float* __restrict__ x7, const float* __restrict__ W,
                       const float* __restrict__ b, float* __restrict__ y, int tokens) {
  int gid = blockIdx.x * 256 + threadIdx.x;
  int tok = gid >> 6, c = gid & 63;
  if (tok >= tokens) return;
  float s = b[c];
  const float* xr = x7 + (size_t)tok * 7;
  for (int i = 0; i < 7; ++i) s += xr[i] * W[i * 64 + c];
  y[(size_t)tok * 64 + c] = s;
}

// ===========================================================================
// 4. Strided/batched WMMA GEMM: C = A(MxK) @ B(KxN) [+bias][+res]
//    MT m-tiles x NT n-tiles per wave; depth-1 software pipeline.
//    PB: B is fragment-packed (pair-rows, ldb = N), else plain row-major.
// ===========================================================================
template <int MT, int NT, bool PB>
__global__ void __launch_bounds__(32)
k_gemm(const float* __restrict__ A, int lda,
       const float* __restrict__ B, int ldb,
       const float* __restrict__ bias,
       const float* __restrict__ res, int ldr,
       float* __restrict__ Cout, int ldc,
       int M, int N, int K, int Z2,
       long long sa1, long long sa2, long long sb1, long long sb2,
       long long sc1, long long sc2) {
  const int lane = threadIdx.x;
  const int r = lane & 15, hi = lane >> 4;
  const int m0 = blockIdx.x * (MT * 16);
  const int n0 = blockIdx.y * (NT * 16);
  const int z = blockIdx.z, z1 = z / Z2, z2 = z % Z2;
  A    += (long long)z1 * sa1 + (long long)z2 * sa2;
  B    += (long long)z1 * sb1 + (long long)z2 * sb2;
  Cout += (long long)z1 * sc1 + (long long)z2 * sc2;

  const v8f vz = {0.f,0.f,0.f,0.f,0.f,0.f,0.f,0.f};
  v8f acc[MT][NT];
#pragma unroll
  for (int mt = 0; mt < MT; ++mt)
#pragma unroll
    for (int t = 0; t < NT; ++t) acc[mt][t] = vz;

  const float* Arow[MT];
#pragma unroll
  for (int mt = 0; mt < MT; ++mt)
    Arow[mt] = A + (size_t)(m0 + mt * 16 + r) * lda + 2 * hi;
  // B fragment base for this lane
  const float* Bb = PB ? (B + ((size_t)hi * ldb + n0 + r) * 2)
                       : (B + (size_t)(2 * hi) * ldb + n0 + r);

  v2f aC[MT], bC[NT];
#pragma unroll
  for (int mt = 0; mt < MT; ++mt) aC[mt] = *(const v2f*)(Arow[mt]);
#pragma unroll
  for (int t = 0; t < NT; ++t) {
    if (PB) bC[t] = *(const v2f*)(Bb + (size_t)t * 32);
    else { const float* p = Bb + t * 16; v2f b; b.x = p[0]; b.y = p[ldb]; bC[t] = b; }
  }

  for (int k0 = 0; k0 < K; k0 += 4) {
    const int kn = (k0 + 4 < K) ? (k0 + 4) : k0;
    v2f aN[MT], bN[NT];
#pragma unroll
    for (int mt = 0; mt < MT; ++mt) aN[mt] = *(const v2f*)(Arow[mt] + kn);
#pragma unroll
    for (int t = 0; t < NT; ++t) {
      if (PB) bN[t] = *(const v2f*)(Bb + ((size_t)(kn >> 1) * ldb + t * 16) * 2);
      else {
        const float* p = Bb + (size_t)kn * ldb + t * 16;
        v2f b; b.x = p[0]; b.y = p[ldb]; bN[t] = b;
      }
    }
#pragma unroll
    for (int t = 0; t < NT; ++t)
#pragma unroll
      for (int mt = 0; mt < MT; ++mt)
        acc[mt][t] = wmma4(aC[mt], bC[t], acc[mt][t]);
#pragma unroll
    for (int mt = 0; mt < MT; ++mt) aC[mt] = aN[mt];
#pragma unroll
    for (int t = 0; t < NT; ++t) bC[t] = bN[t];
  }

#pragma unroll
  for (int mt = 0; mt < MT; ++mt)
#pragma unroll
    for (int t = 0; t < NT; ++t) {
      int nn = n0 + t * 16 + r;
      float bs = bias ? bias[nn] : 0.f;
#pragma unroll
      for (int rr = 0; rr < 8; ++rr) {
        int row = m0 + mt * 16 + rr + 8 * hi;
        float v = acc[mt][t][rr] + bs;
        if (res) v += res[(size_t)row * ldr + nn];
        Cout[(size_t)row * ldc + nn] = v;
      }
    }
}

// ===========================================================================
// 5. Fused MLP: out = [res +] gelu(x@W1+b1)@W2+b2.   x: Mx64,
//    W1p: packed 64x256, W2p: packed 256xN2 (N2 = 64 or 3).
// ===========================================================================
template <int N2>
__global__ void __launch_bounds__(32)
k_mlp(const float* __restrict__ x,
      const float* __restrict__ W1p, const float* __restrict__ b1,
      const float* __restrict__ W2p, const float* __restrict__ b2,
      const float* __restrict__ res, float* __restrict__ out) {
  __shared__ float hl[16 * 264];
  const int lane = threadIdx.x;
  const int r = lane & 15, hi = lane >> 4;
  const size_t m0 = (size_t)blockIdx.x * 16;
  const v8f vz = {0.f,0.f,0.f,0.f,0.f,0.f,0.f,0.f};

  v8f hid[16];
#pragma unroll
  for (int t = 0; t < 16; ++t) hid[t] = vz;

  const float* Arow = x + (m0 + r) * 64 + 2 * hi;
  const float* B1 = W1p + ((size_t)hi * 256 + r) * 2;

  v2f aC = *(const v2f*)(Arow);
  v2f bC[16];
#pragma unroll
  for (int t = 0; t < 16; ++t) bC[t] = *(const v2f*)(B1 + (size_t)t * 32);

  for (int k0 = 0; k0 < 64; k0 += 4) {
    const int kn = (k0 + 4 < 64) ? (k0 + 4) : k0;
    v2f aN = *(const v2f*)(Arow + kn);
    v2f bN[16];
#pragma unroll
    for (int t = 0; t < 16; ++t)
      bN[t] = *(const v2f*)(B1 + ((size_t)(kn >> 1) * 256 + t * 16) * 2);
#pragma unroll
    for (int t = 0; t < 16; ++t) hid[t] = wmma4(aC, bC[t], hid[t]);
    aC = aN;
#pragma unroll
    for (int t = 0; t < 16; ++t) bC[t] = bN[t];
  }
  // bias + gelu -> LDS (row-major 16 x 256, padded to 264)
#pragma unroll
  for (int t = 0; t < 16; ++t) {
    int col = t * 16 + r;
    float bb = b1[col];
#pragma unroll
    for (int rr = 0; rr < 8; ++rr) {
      float v = hid[t][rr] + bb;
      hl[(rr + 8 * hi) * 264 + col] = gelu_tanh(v);
    }
  }
  __syncthreads();

  constexpr int ONT = (N2 + 15) / 16;
  v8f o4[ONT];
#pragma unroll
  for (int t = 0; t < ONT; ++t) o4[t] = vz;

  const float* Hrow = hl + r * 264 + 2 * hi;
  const float* B2 = W2p + (size_t)hi * N2 * 2;

  v2f a2C = *(const v2f*)(Hrow);
  v2f b2C[ONT];
#pragma unroll
  for (int t = 0; t < ONT; ++t) {
    int col = t * 16 + r;
    v2f b = {0.f, 0.f};
    if (col < N2) b = *(const v2f*)(B2 + (size_t)col * 2);
    b2C[t] = b;
  }

  for (int k0 = 0; k0 < 256; k0 += 4) {
    const int kn = (k0 + 4 < 256) ? (k0 + 4) : k0;
    v2f a2N = *(const v2f*)(Hrow + kn);
    v2f b2N[ONT];
#pragma unroll
    for (int t = 0; t < ONT; ++t) {
      int col = t * 16 + r;
      v2f b = {0.f, 0.f};
      if (col < N2) b = *(const v2f*)(B2 + ((size_t)(kn >> 1) * N2 + col) * 2);
      b2N[t] = b;
    }
#pragma unroll
    for (int t = 0; t < ONT; ++t) o4[t] = wmma4(a2C, b2C[t], o4[t]);
    a2C = a2N;
#pragma unroll
    for (int t = 0; t < ONT; ++t) b2C[t] = b2N[t];
  }
#pragma unroll
  for (int t = 0; t < ONT; ++t) {
    int col = t * 16 + r;
    if (col < N2) {
      float bb = b2[col];
#pragma unroll
      for (int rr = 0; rr < 8; ++rr) {
        size_t row = m0 + rr + 8 * hi;
        float v = o4[t][rr] + bb;
        if (res) v += res[row * N2 + col];
        out[row * N2 + col] = v;
      }
    }
  }
}

// ===========================================================================
// 6. Attention helpers
// ===========================================================================
__global__ void k_qkmean(const float* __restrict__ qkv, float* __restrict__ qm,
                         float* __restrict__ km) {
  int e = blockIdx.x, c = threadIdx.x;     // 512 blocks x 128 threads
  float sq = 0.f, sk = 0.f;
  for (int p = 0; p < 64; ++p) {
    const float* row = qkv + ((size_t)e * 64 + p) * 384;
    sq += row[c];
    sk += row[128 + c];
  }
  qm[e * 128 + c] = sq * (1.f / 64.f);
  km[e * 128 + c] = sk * (1.f / 64.f);
}

__global__ void k_attnsoft(const float* __restrict__ qm, const float* __restrict__ km,
                           float* __restrict__ attn) {
  __shared__ float sv[512];
  __shared__ float red[512];
  int e = blockIdx.x, h = blockIdx.y, f = threadIdx.x;  // 512 threads
  const float* q = qm + e * 128 + h * 32;
  const float* kk = km + f * 128 + h * 32;
  float s = 0.f;
  for (int d = 0; d < 32; ++d) s += q[d] * kk[d];
  s *= 0.17677669529663687f;    // 1/sqrt(32)
  sv[f] = s; red[f] = s;
  __syncthreads();
  for (int st = 256; st > 0; st >>= 1) {
    if (f < st) red[f] = fmaxf(red[f], red[f + st]);
    __syncthreads();
  }
  float mx = red[0];
  __syncthreads();
  float ex = expf(sv[f] - mx);
  red[f] = ex;
  __syncthreads();
  for (int st = 256; st > 0; st >>= 1) {
    if (f < st) red[f] += red[f + st];
    __syncthreads();
  }
  attn[((size_t)h * 512 + e) * 512 + f] = ex / red[0];
}

// ===========================================================================
// 7. 27-point periodic element stencil
// ===========================================================================
__global__ void k_stencil(const float* __restrict__ m, const float* __restrict__ w,
                          float* __restrict__ acc, int P3) {
  int e = blockIdx.x;
  int mi = e >> 6, mj = (e >> 3) & 7, mk = e & 7;
  int total = P3 * 64;
  for (int i = threadIdx.x; i < total; i += blockDim.x) {
    float s = 0.f;
    int t = 0;
    for (int di = -1; di <= 1; ++di)
      for (int dj = -1; dj <= 1; ++dj)
        for (int dk = -1; dk <= 1; ++dk) {
          int ne = (((mi - di) & 7) << 6) | (((mj - dj) & 7) << 3) | ((mk - dk) & 7);
          s += w[(size_t)t * total + i] * m[(size_t)ne * total + i];
          ++t;
        }
    acc[(size_t)e * total + i] = s;
  }
}

// ===========================================================================
// 8. Unpack (e, pt, c) -> (X, Y, Z, c)
// ===========================================================================
__global__ void k_unpack(const float* __restrict__ xe, float* __restrict__ out) {
  int gid = blockIdx.x * 256 + threadIdx.x;
  if (gid >= 512 * 512 * 3) return;
  int c = gid % 3, rest = gid / 3;
  int pt = rest % 512, e = rest / 512;
  int mi = e >> 6, mj = (e >> 3) & 7, mk = e & 7;
  int i = pt >> 6, j = (pt >> 3) & 7, k = pt & 7;
  out[(((size_t)(mi*8+i) * 64 + (mj*8+j)) * 64 + (mk*8+k)) * 3 + c] = xe[gid];
}

// ===========================================================================
// Host launch
// ===========================================================================
extern "C" void kernel_launch(void* const* d_in, const int* in_sizes, int n_in,
                              void* d_out, int out_size, void* d_ws, size_t ws_size,
                              hipStream_t stream) {
  (void)in_sizes; (void)n_in; (void)out_size; (void)ws_size;
  // --- inputs in jax pytree (sorted-key) flatten order ---
  const float* input = (const float*)d_in[0];
  const float* aWo   = (const float*)d_in[1];   // (2,128,64)
  const float* aWqkv = (const float*)d_in[2];   // (2,64,384)
  const float* abo   = (const float*)d_in[3];   // (2,64)
  const float* abqkv = (const float*)d_in[4];   // (2,384)
  const float* clWm  = (const float*)d_in[5];   // (2,64,64)
  const float* clbm  = (const float*)d_in[6];   // (2,64)
  const float* clw   = (const float*)d_in[7];   // (2,27,4,4,4,64)
  const float* csWm  = (const float*)d_in[8];   // (2,64,64)
  const float* csbm  = (const float*)d_in[9];   // (2,64)
  const float* csw   = (const float*)d_in[10];  // (2,27,8,8,8,64)
  const float* eps   = (const float*)d_in[11];  // (2,64)
  const float* flW1  = (const float*)d_in[12];  // (2,64,256)
  const float* flW2  = (const float*)d_in[13];  // (2,256,64)
  const float* flb1  = (const float*)d_in[14];
  const float* flb2  = (const float*)d_in[15];
  const float* fsW1  = (const float*)d_in[16];
  const float* fsW2  = (const float*)d_in[17];
  const float* fsb1  = (const float*)d_in[18];
  const float* fsb2  = (const float*)d_in[19];
  const float* llW   = (const float*)d_in[20];  // lift_les W (7,64)
  const float* llb   = (const float*)d_in[21];
  const float* lsW   = (const float*)d_in[22];  // lift_sgs W (7,64)
  const float* lsb   = (const float*)d_in[23];
  const float* plW1  = (const float*)d_in[24];
  const float* plb1  = (const float*)d_in[25];
  const float* plW2  = (const float*)d_in[26];  // (256,3)
  const float* plb2  = (const float*)d_in[27];
  const float* psW1  = (const float*)d_in[28];
  const float* psb1  = (const float*)d_in[29];
  const float* psW2  = (const float*)d_in[30];
  const float* psb2  = (const float*)d_in[31];

  // --- workspace layout (floats) ---
  float* ws   = (float*)d_ws;
  float* MATS = ws;
  float* XG7  = MATS + MATS_F;                            // 512*512*7
  float* LES7 = XG7 + (size_t)NE * P8_3 * 7;              // 512*64*7
  float* SGS  = LES7 + (size_t)NE * P4_3 * 7;             // 262144*64
  float* LES  = SGS + (size_t)TOK_S * CCH;                // 32768*64
  float* T1   = LES + (size_t)TOK_L * CCH;                // 262144*64
  float* T2   = T1 + (size_t)TOK_S * CCH;                 // 262144*64
  float* ATT  = T2 + (size_t)TOK_S * CCH;                 // 4*512*512
  float* QM   = ATT + (size_t)NHEAD * NE * NE;            // 512*128
  float* KM   = QM + NE * 128;                            // 512*128
  // fragment-packed weights (same sizes as originals)
  float* P    = KM + NE * 128;
  float* pWqkv = P;            P += (size_t)NLAY * 64 * 384;
  float* pWo   = P;            P += (size_t)NLAY * 128 * 64;
  float* pClWm = P;            P += (size_t)NLAY * 64 * 64;
  float* pCsWm = P;            P += (size_t)NLAY * 64 * 64;
  float* pFlW1 = P;            P += (size_t)NLAY * 64 * 256;
  float* pFlW2 = P;            P += (size_t)NLAY * 256 * 64;
  float* pFsW1 = P;            P += (size_t)NLAY * 64 * 256;
  float* pFsW2 = P;            P += (size_t)NLAY * 256 * 64;
  float* pPlW1 = P;            P += (size_t)64 * 256;
  float* pPlW2 = P;            P += (size_t)256 * 3;
  float* pPsW1 = P;            P += (size_t)64 * 256;
  float* pPsW2 = P;            P += (size_t)256 * 3;

  k_init_mats<<<1, 32, 0, stream>>>(MATS);

  // --- pack all GEMM weights into fragment order ---
  for (int l = 0; l < NLAY; ++l) {
    k_packB<<<(32*384 + 255)/256, 256, 0, stream>>>(aWqkv + (size_t)l*64*384, pWqkv + (size_t)l*64*384, 64, 384);
    k_packB<<<(64*64  + 255)/256, 256, 0, stream>>>(aWo   + (size_t)l*128*64, pWo   + (size_t)l*128*64, 128, 64);
    k_packB<<<(32*64  + 255)/256, 256, 0, stream>>>(clWm  + (size_t)l*64*64,  pClWm + (size_t)l*64*64,  64, 64);
    k_packB<<<(32*64  + 255)/256, 256, 0, stream>>>(csWm  + (size_t)l*64*64,  pCsWm + (size_t)l*64*64,  64, 64);
    k_packB<<<(32*256 + 255)/256, 256, 0, stream>>>(flW1  + (size_t)l*64*256, pFlW1 + (size_t)l*64*256, 64, 256);
    k_packB<<<(128*64 + 255)/256, 256, 0, stream>>>(flW2  + (size_t)l*256*64, pFlW2 + (size_t)l*256*64, 256, 64);
    k_packB<<<(32*256 + 255)/256, 256, 0, stream>>>(fsW1  + (size_t)l*64*256, pFsW1 + (size_t)l*64*256, 64, 256);
    k_packB<<<(128*64 + 255)/256, 256, 0, stream>>>(fsW2  + (size_t)l*256*64, pFsW2 + (size_t)l*256*64, 256, 64);
  }
  k_packB<<<(32*256 + 255)/256, 256, 0, stream>>>(plW1, pPlW1, 64, 256);
  k_packB<<<(128*3  + 255)/256, 256, 0, stream>>>(plW2, pPlW2, 256, 3);
  k_packB<<<(32*256 + 255)/256, 256, 0, stream>>>(psW1, pPsW1, 64, 256);
  k_packB<<<(128*3  + 255)/256, 256, 0, stream>>>(psW2, pPsW2, 256, 3);

  k_embed<<<NE, 256, 0, stream>>>(input, XG7, MATS);
  k_apply3<<<NE, 256, 0, stream>>>(XG7, LES7, MATS + MOFF_DN, 8, 4, 7, nullptr, 0);
  k_lift<<<(TOK_S * 64) / 256, 256, 0, stream>>>(XG7, lsW, lsb, SGS, TOK_S);
  k_lift<<<(TOK_L * 64) / 256, 256, 0, stream>>>(LES7, llW, llb, LES, TOK_L);

  for (int l = 0; l < NLAY; ++l) {
    const float* bqkv = abqkv + (size_t)l * 384;
    const float* bo   = abo   + (size_t)l * 64;

    // --- attention on LES ---
    // qkv = les @ Wqkv + bqkv        (32768 x 384)
    k_gemm<2,4,true><<<dim3(TOK_L/32, 6, 1), 32, 0, stream>>>(
        LES, 64, pWqkv + (size_t)l*64*384, 384, bqkv, nullptr, 0, T1, 384,
        TOK_L, 384, 64, 1, 0, 0, 0, 0, 0, 0);
    k_qkmean<<<NE, 128, 0, stream>>>(T1, QM, KM);
    k_attnsoft<<<dim3(NE, NHEAD), 512, 0, stream>>>(QM, KM, ATT);
    // o[e,p,h,d] = sum_f attn[h,e,f] * v[f,p,h,d]   (batched over z=(h,p))
    k_gemm<2,2,false><<<dim3(NE/32, 1, NHEAD * 64), 32, 0, stream>>>(
        ATT, NE, T1 + 256, 64 * 384, nullptr, nullptr, 0, T2, 64 * 128,
        NE, 32, NE, 64,
        (long long)NE * NE, 0,          // A: h stride, p stride
        32, 384,                        // B: h stride, p stride
        32, 128);                       // C: h stride, p stride
    // les += o @ Wo + bo
    k_gemm<2,4,true><<<dim3(TOK_L/32, 1, 1), 32, 0, stream>>>(
        T2, 128, pWo + (size_t)l*128*64, 64, bo, LES, 64, LES, 64,
        TOK_L, 64, 128, 1, 0, 0, 0, 0, 0, 0);

    // --- sem_conv + ffn on LES ---
    k_apply3<<<NE, 256, 0, stream>>>(LES, T1, MATS + MOFF_C2M4, 4, 4, CCH, nullptr, 0);
    k_stencil<<<NE, 256, 0, stream>>>(T1, clw + (size_t)l * 27 * P4_3 * CCH, T2, P4_3);
    k_apply3<<<NE, 256, 0, stream>>>(T2, T1, MATS + MOFF_M2N4, 4, 4, CCH, nullptr, 0);
    k_gemm<2,4,true><<<dim3(TOK_L/32, 1, 1), 32, 0, stream>>>(
        T1, 64, pClWm + (size_t)l*64*64, 64, clbm + (size_t)l * 64,
        nullptr, 0, T2, 64, TOK_L, 64, 64, 1, 0, 0, 0, 0, 0, 0);
    k_mlp<64><<<TOK_L/16, 32, 0, stream>>>(
        T2, pFlW1 + (size_t)l*64*256, flb1 + (size_t)l * 256,
        pFlW2 + (size_t)l*256*64, flb2 + (size_t)l * 64, LES, LES);

    // --- sgs += eps * up(les) ---
    k_apply3<<<NE, 256, 0, stream>>>(LES, SGS, MATS + MOFF_UP, 4, 8, CCH,
                                     eps + (size_t)l * 64, 1);

    // --- sem_conv + ffn on SGS ---
    k_apply3<<<NE, 256, 0, stream>>>(SGS, T1, MATS + MOFF_C2M8, 8, 8, CCH, nullptr, 0);
    k_stencil<<<NE, 256, 0, stream>>>(T1, csw + (size_t)l * 27 * P8_3 * CCH, T2, P8_3);
    k_apply3<<<NE, 256, 0, stream>>>(T2, T1, MATS + MOFF_M2N8, 8, 8, CCH, nullptr, 0);
    k_gemm<2,4,true><<<dim3(TOK_S/32, 1, 1), 32, 0, stream>>>(
        T1, 64, pCsWm + (size_t)l*64*64, 64, csbm + (size_t)l * 64,
        nullptr, 0, T2, 64, TOK_S, 64, 64, 1, 0, 0, 0, 0, 0, 0);
    k_mlp<64><<<TOK_S/16, 32, 0, stream>>>(
        T2, pFsW1 + (size_t)l*64*256, fsb1 + (size_t)l * 256,
        pFsW2 + (size_t)l*256*64, fsb2 + (size_t)l * 64, SGS, SGS);
  }

  // --- projections, upsample + evaluate, unpack ---
  k_mlp<3><<<TOK_L/16, 32, 0, stream>>>(LES, pPlW1, plb1, pPlW2, plb2, nullptr, T1);
  k_mlp<3><<<TOK_S/16, 32, 0, stream>>>(SGS, pPsW1, psb1, pPsW2, psb2, nullptr, T2);
  k_apply3<<<NE, 256, 0, stream>>>(T1, T2, MATS + MOFF_UP, 4, 8, 3, nullptr, 1);
  k_apply3<<<NE, 256, 0, stream>>>(T2, T1, MATS + MOFF_EV, 8, 8, 3, nullptr, 0);
  k_unpack<<<(512 * 512 * 3 + 255) / 256, 256, 0, stream>>>(T1, (float*)d_out);
}